// GlacierDynamicsCheckpointed_39659728011549
// MI455X (gfx1250) — compile-verified
//
#include <hip/hip_runtime.h>
#include <stdint.h>

// ---------------- problem constants (match reference) ----------------
#define NYg 1024
#define NXg 1024
static __device__ __constant__ const float kDX    = 50.0f;
static __device__ __constant__ const float kInvDX = 1.0f / 50.0f;
static __device__ __constant__ const float kTTOT  = 32.0f;
static __device__ __constant__ const float kDTMAX = 0.5f;
static __device__ __constant__ const float kEPS   = 1e-10f;
static __device__ __constant__ const float kTFREQ = 5.0f;
static __device__ __constant__ const float kLAPSE = 0.0065f;
static __device__ __constant__ const float kMELT  = 0.5f;
// FD * (RHO*G)^3 = 1e-17 * (910*9.81)^3
static __device__ __constant__ const float kCDIFF =
    (float)(1e-17 * (910.0 * 9.81) * (910.0 * 9.81) * (910.0 * 9.81));

#define N_STEPS 64
#define BXs 32
#define BYs 8

struct Scalars {
    float time;
    float t_last;
    float dt;
    int   flag;       // refresh SMB this step?
    unsigned maxBits; // bit-pattern of running max(D) (all D > 0)
};

// ---------------- CDNA5 async global->LDS copy helpers ----------------
// ASYNCcnt-tracked DMA-style per-lane copies into LDS (no VGPR staging).
// LDS byte offset = low 32 bits of the generic shared-aperture address
// (aperture lives entirely in addr[63:32] per CDNA5 ISA §10.2).
__device__ __forceinline__ uint32_t lds_off_of(const void* p) {
    return (uint32_t)(uintptr_t)p;
}
__device__ __forceinline__ void async_ld_b128(uint32_t lds_off, const float* gptr) {
    uint64_t ga = (uint64_t)(uintptr_t)gptr;
    asm volatile("global_load_async_to_lds_b128 %0, %1, off"
                 :: "v"(lds_off), "v"(ga)
                 : "memory");
}
__device__ __forceinline__ void wait_async0() {
    asm volatile("s_wait_asynccnt 0" ::: "memory");
}

// ---------------- kernel 0: init state ----------------
__global__ __launch_bounds__(256) void k_init(
    float* __restrict__ H, float* __restrict__ Z0, float* __restrict__ smb,
    const float* __restrict__ Ztopo, const float* __restrict__ precip,
    const float* __restrict__ Tma, const float* __restrict__ Tmj,
    const float* __restrict__ mask, Scalars* sc)
{
    const int j = blockIdx.x * BXs + threadIdx.x;
    const int i = blockIdx.y * BYs + threadIdx.y;
    const size_t idx = (size_t)i * NXg + j;
    const float z = Ztopo[idx];
    H[idx]  = 0.0f;
    Z0[idx] = z;
    const float tma = *Tma - kLAPSE * z;
    const float tmj = *Tmj - kLAPSE * z;
    const float acc = (tma < 0.0f) ? precip[idx] : 0.0f;
    smb[idx] = (acc - kMELT * fmaxf(tmj, 0.0f)) * mask[idx];
    if (i == 0 && j == 0) {
        sc->time = 0.0f; sc->t_last = 0.0f; sc->dt = 0.0f;
        sc->flag = 0;    sc->maxBits = 0u;
    }
}

// ---------------- kernel 1: staggered diffusivity D + max(D) ----------------
// Tile covers gi = r0..r0+8 (clamped <=1023), gj = c0..c0+35 (chunk-clamped).
// cc <-> gj = c0+cc. Only cc<=32 / rr<=8 slots feed in-domain threads; every
// clamped duplicate slot is read solely by out-of-domain (guarded) threads.
__global__ __launch_bounds__(256) void k_computeD(
    const float* __restrict__ H, const float* __restrict__ Z,
    float* __restrict__ D, unsigned* __restrict__ maxBits)
{
    __shared__ __align__(16) float sH[BYs + 1][36];   // 9 x 36 (144B pitch)
    __shared__ __align__(16) float sZ[BYs + 1][36];
    __shared__ float sWave[256 / 32];

    const int tx = threadIdx.x, ty = threadIdx.y;
    const int tid = ty * BXs + tx;
    const int c0 = blockIdx.x * BXs, r0 = blockIdx.y * BYs;

    // 9 rows x 9 x B128 chunks per array; one chunk per thread, no loop.
    if (tid < 81) {
        const int rr = tid / 9, c = tid % 9;
        const int gi = min(r0 + rr, NYg - 1);
        const int gj = min(c0 + 4 * c, NXg - 4);      // 16B-aligned base
        const size_t g = (size_t)gi * NXg + gj;
        async_ld_b128(lds_off_of(&sH[rr][4 * c]), H + g);
        async_ld_b128(lds_off_of(&sZ[rr][4 * c]), Z + g);
    }
    wait_async0();
    __syncthreads();

    const int i = r0 + ty, j = c0 + tx;
    float d = 0.0f;
    if (i < NYg - 1 && j < NXg - 1) {
        const float h00 = sH[ty][tx],     h01 = sH[ty][tx + 1];
        const float h10 = sH[ty + 1][tx], h11 = sH[ty + 1][tx + 1];
        const float z00 = sZ[ty][tx],     z01 = sZ[ty][tx + 1];
        const float z10 = sZ[ty + 1][tx], z11 = sZ[ty + 1][tx + 1];
        const float havg = 0.25f * (h00 + h01 + h10 + h11);
        const float sx = 0.5f * ((z01 - z00) + (z11 - z10)) * kInvDX;
        const float sy = 0.5f * ((z10 - z00) + (z11 - z01)) * kInvDX;
        const float s2 = sx * sx + sy * sy + kEPS;    // == Snorm^2
        const float h2 = havg * havg, h4 = h2 * h2;
        d = kCDIFF * h4 * havg * s2 + kEPS;
        D[(size_t)i * NXg + j] = d;                   // pitch NXg (padded)
    }
    // wave32 max reduce, then 1 atomic per block (bit order == float order, D>0)
    float m = d;
    for (int off = 16; off > 0; off >>= 1) m = fmaxf(m, __shfl_xor(m, off));
    if ((tid & 31) == 0) sWave[tid >> 5] = m;
    __syncthreads();
    if (tid < 256 / 32) {
        float w = sWave[tid];
        for (int off = 4; off > 0; off >>= 1) w = fmaxf(w, __shfl_xor(w, off));
        if (tid == 0) atomicMax(maxBits, __float_as_uint(w));
    }
}

// ---------------- kernel 2: adaptive CFL dt (serial point) ----------------
__global__ void k_dt(Scalars* sc)
{
    const float maxD = __uint_as_float(sc->maxBits);
    float dt = fminf((kDX * kDX) / (2.7f * maxD), kDTMAX);
    if (sc->time >= kTTOT) dt = 0.0f;        // freeze after ttot
    const float t = sc->time + dt;
    const int flag = (t - sc->t_last) >= kTFREQ;
    if (flag) sc->t_last = t;
    sc->time = t;
    sc->dt = dt;
    sc->flag = flag;
    sc->maxBits = 0u;                        // reset for next step
}

// ---------------- kernel 3: flux divergence + H/Z/SMB update ----------------
// Z tile: gi = r0-1..r0+8, cc <-> gj = c0-4+cc (cc=0..39), zc at [ty+1][tx+4].
// D tile: gi = r0-1..r0+7, cc <-> gj = c0-4+cc (cc=0..35), d00 at [ty][tx+3].
// Edge-clamped chunks only land in slots read by border threads (discarded).
__global__ __launch_bounds__(256) void k_update(
    float* __restrict__ H, const float* __restrict__ Zin,
    float* __restrict__ Zout, const float* __restrict__ D,
    float* __restrict__ smb, const float* __restrict__ Ztopo,
    const float* __restrict__ precip, const float* __restrict__ Tma,
    const float* __restrict__ Tmj, const float* __restrict__ mask,
    const Scalars* __restrict__ sc)
{
    __shared__ __align__(16) float sZ[BYs + 2][40];   // 10 x 40 (160B pitch)
    __shared__ __align__(16) float sD[BYs + 1][36];   // 9 x 36  (144B pitch)

    const int tx = threadIdx.x, ty = threadIdx.y;
    const int tid = ty * BXs + tx;
    const int c0 = blockIdx.x * BXs, r0 = blockIdx.y * BYs;

    if (tid < 100) {                                  // 10 rows x 10 B128 chunks
        const int rr = tid / 10, c = tid % 10;
        const int gi = min(max(r0 - 1 + rr, 0), NYg - 1);
        const int gj = min(max(c0 - 4 + 4 * c, 0), NXg - 4);
        async_ld_b128(lds_off_of(&sZ[rr][4 * c]), Zin + (size_t)gi * NXg + gj);
    }
    if (tid < 81) {                                   // 9 rows x 9 B128 chunks
        const int rr = tid / 9, c = tid % 9;
        const int gi = min(max(r0 - 1 + rr, 0), NYg - 2);
        const int gj = min(max(c0 - 4 + 4 * c, 0), NXg - 4);
        async_ld_b128(lds_off_of(&sD[rr][4 * c]), D + (size_t)gi * NXg + gj);
    }
    wait_async0();
    __syncthreads();

    const int i = r0 + ty, j = c0 + tx;
    const size_t idx = (size_t)i * NXg + j;
    const float dt = sc->dt;
    const int flag = sc->flag;

    const float h = H[idx];
    float hn = h;
    if (i >= 1 && i <= NYg - 2 && j >= 1 && j <= NXg - 2) {
        const float zc = sZ[ty + 1][tx + 4];
        const float zl = sZ[ty + 1][tx + 3], zr = sZ[ty + 1][tx + 5];
        const float zu = sZ[ty][tx + 4],     zd = sZ[ty + 2][tx + 4];
        const float d00 = sD[ty][tx + 3],    d01 = sD[ty][tx + 4];
        const float d10 = sD[ty + 1][tx + 3], d11 = sD[ty + 1][tx + 4];
        // qx[a][b+1], qx[a][b], qy[a+1][b], qy[a][b]  (a=i-1, b=j-1)
        const float qxR = -0.5f * (d01 + d11) * (zr - zc) * kInvDX;
        const float qxL = -0.5f * (d00 + d10) * (zc - zl) * kInvDX;
        const float qyD = -0.5f * (d10 + d11) * (zd - zc) * kInvDX;
        const float qyU = -0.5f * (d00 + d01) * (zc - zu) * kInvDX;
        const float dHdt = -((qxR - qxL) * kInvDX + (qyD - qyU) * kInvDX);
        hn = fmaxf(h + dt * (dHdt + smb[idx]), 0.0f);
    }
    H[idx] = hn;                     // border rows keep H (=0) forever
    const float zn = Ztopo[idx] + hn;
    Zout[idx] = zn;
    if (flag) {                      // SMB refresh every T_FREQ model-years
        const float tma = *Tma - kLAPSE * zn;
        const float tmj = *Tmj - kLAPSE * zn;
        const float acc = (tma < 0.0f) ? precip[idx] : 0.0f;
        smb[idx] = (acc - kMELT * fmaxf(tmj, 0.0f)) * mask[idx];
    }
}

// ---------------- host-side orchestration ----------------
extern "C" void kernel_launch(void* const* d_in, const int* in_sizes, int n_in,
                              void* d_out, int out_size, void* d_ws, size_t ws_size,
                              hipStream_t stream) {
    (void)in_sizes; (void)n_in; (void)out_size; (void)ws_size;
    const float* precip = (const float*)d_in[0];
    const float* Tma    = (const float*)d_in[1];
    const float* Tmj    = (const float*)d_in[2];
    const float* Ztopo  = (const float*)d_in[3];
    const float* mask   = (const float*)d_in[4];

    float* H = (float*)d_out;                 // final H lives in d_out directly
    const size_t N = (size_t)NYg * NXg;
    float* ws  = (float*)d_ws;
    float* Z0  = ws;                          // double-buffered surface
    float* Z1  = ws + N;
    float* smb = ws + 2 * N;
    float* D   = ws + 3 * N;                  // padded to NXg pitch
    Scalars* sc = (Scalars*)(ws + 4 * N);

    const dim3 blk(BXs, BYs);                 // 256 threads = 8 wave32
    const dim3 grd(NXg / BXs, NYg / BYs);     // 32 x 128

    k_init<<<grd, blk, 0, stream>>>(H, Z0, smb, Ztopo, precip, Tma, Tmj, mask, sc);

    float* zin = Z0;
    float* zout = Z1;
    for (int s = 0; s < N_STEPS; ++s) {
        k_computeD<<<grd, blk, 0, stream>>>(H, zin, D, &sc->maxBits);
        k_dt<<<1, 1, 0, stream>>>(sc);
        k_update<<<grd, blk, 0, stream>>>(H, zin, zout, D, smb, Ztopo,
                                          precip, Tma, Tmj, mask, sc);
        float* t = zin; zin = zout; zout = t;
    }
}